// DGCNN_DE_46291157516701
// MI455X (gfx1250) — compile-verified
//
#include <hip/hip_runtime.h>
#include <hip/hip_bf16.h>
#include <math.h>

#define ALPHA_LR 0.2f
#define NEG_INF -1.0e12f

constexpr int B = 128, N = 512, F_IN = 64, H = 8, F_HID = 32, C = 64, HF = H * F_HID;

typedef float v2f __attribute__((ext_vector_type(2)));
typedef float v8f __attribute__((ext_vector_type(8)));

__device__ __forceinline__ v8f wmma_f32_k4(v2f a, v2f b, v8f c) {
  // D = A(16x4,f32) * B(4x16,f32) + C(16x16,f32)
  return __builtin_amdgcn_wmma_f32_16x16x4_f32(false, a, false, b, (short)0, c, false, false);
}
__device__ __forceinline__ float lrelu(float x) { return x > 0.f ? x : ALPHA_LR * x; }

// -------- mask[i*N+j] = (A[i][j] > 0); equivalent to (L>0) since dinv>0 --------
__global__ void k_mask(const float* __restrict__ A, unsigned char* __restrict__ mask) {
  int idx = blockIdx.x * blockDim.x + threadIdx.x;
  if (idx < N * N) mask[idx] = (A[idx] > 0.f) ? 1 : 0;
}

// -------- h[b,hh,n,o] = x[b,n,:] @ W[hh,:,o]   (one wave = 16 rows x 32 cols) --------
__global__ void __launch_bounds__(256) k_h(const float* __restrict__ x,
                                           const float* __restrict__ W,
                                           float* __restrict__ h) {
  const int lane = threadIdx.x & 31;
  const int wave = blockIdx.x * (blockDim.x >> 5) + (threadIdx.x >> 5);
  const int rt = wave & 31;          // 32 row tiles of 16
  const int bh = wave >> 5;          // b*H + hh
  const int hh = bh & (H - 1);
  const int b  = bh >> 3;
  const int i0 = rt * 16;
  const int half = lane >> 4;
  const int l = lane & 15;

  const float* xr = x + ((size_t)b * N + i0 + l) * F_IN;
  const float* Wh = W + (size_t)hh * F_IN * F_HID;
  v8f c0 = {}, c1 = {};
  for (int k = 0; k < F_IN; k += 4) {
    const int kk = k + 2 * half;
    v2f a; a.x = xr[kk]; a.y = xr[kk + 1];
    const float* w0 = Wh + (size_t)kk * F_HID;
    const float* w1 = Wh + (size_t)(kk + 1) * F_HID;
    v2f b0, b1;
    b0.x = w0[l];      b0.y = w1[l];
    b1.x = w0[16 + l]; b1.y = w1[16 + l];
    c0 = wmma_f32_k4(a, b0, c0);
    c1 = wmma_f32_k4(a, b1, c1);
  }
  float* hb = h + (size_t)bh * N * F_HID;
  for (int r = 0; r < 8; r++) {
    int m = i0 + r + 8 * half;
    hb[(size_t)m * F_HID + l]      = c0[r];
    hb[(size_t)m * F_HID + 16 + l] = c1[r];
  }
}

// -------- e1/e2[b,hh,n] = h[b,hh,n,:] . a1/a2[hh,:] --------
__global__ void k_e(const float* __restrict__ h, const float* __restrict__ a1,
                    const float* __restrict__ a2, float* __restrict__ e1,
                    float* __restrict__ e2) {
  int idx = blockIdx.x * blockDim.x + threadIdx.x;
  if (idx >= B * H * N) return;
  const int hh = (idx / N) & (H - 1);
  const float* p = h + (size_t)idx * F_HID;
  const float* A1 = a1 + hh * F_HID;
  const float* A2 = a2 + hh * F_HID;
  float s1 = 0.f, s2 = 0.f;
  for (int o = 0; o < F_HID; o++) { float v = p[o]; s1 += v * A1[o]; s2 += v * A2[o]; }
  e1[idx] = s1; e2[idx] = s2;
}

// -------- per-column softmax stats: cpk[page,j] = {e2[j], colmax[j], 1/colsum[j], 0} --------
// gridDim.x = #pages (B*H or B), blockDim.x = N
__global__ void k_colstats(const unsigned char* __restrict__ mask,
                           const float* __restrict__ e1, const float* __restrict__ e2,
                           float4* __restrict__ cpk) {
  const int pg = blockIdx.x;
  const int j = threadIdx.x;
  const float* e1p = e1 + (size_t)pg * N;
  const float ej = e2[(size_t)pg * N + j];
  float mx = -3.0e38f;
  for (int i = 0; i < N; i++) {
    float e = mask[i * N + j] ? lrelu(e1p[i] + ej) : NEG_INF;
    mx = fmaxf(mx, e);
  }
  float s = 0.f;
  for (int i = 0; i < N; i++) {
    float e = mask[i * N + j] ? lrelu(e1p[i] + ej) : NEG_INF;
    s += expf(e - mx);
  }
  cpk[(size_t)pg * N + j] = make_float4(ej, mx, 1.f / s, 0.f);
}

// -------- xcat[b,i,hh*32+o] = relu( sum_j P[i,j]*h[b,hh,j,o] ), P built in-flight --------
__global__ void __launch_bounds__(256) k_att(const unsigned char* __restrict__ mask,
                                             const float* __restrict__ e1,
                                             const float4* __restrict__ cpk,
                                             const float* __restrict__ h,
                                             float* __restrict__ xcat) {
  const int lane = threadIdx.x & 31;
  const int wave = blockIdx.x * (blockDim.x >> 5) + (threadIdx.x >> 5);
  const int rt = wave & 31;
  const int bh = wave >> 5;
  const int hh = bh & (H - 1);
  const int b  = bh >> 3;
  const int i0 = rt * 16;
  const int half = lane >> 4;
  const int l = lane & 15;

  const float e1r = e1[(size_t)bh * N + i0 + l];
  const unsigned char* mrow = mask + (size_t)(i0 + l) * N;
  const float4* cp = cpk + (size_t)bh * N;
  const float* hb = h + (size_t)bh * N * F_HID;

  v8f c0 = {}, c1 = {};
  for (int j = 0; j < N; j += 4) {
    const int jk = j + 2 * half;
    float4 q0 = cp[jk];
    float4 q1 = cp[jk + 1];
    float em0 = mrow[jk]     ? lrelu(e1r + q0.x) : NEG_INF;
    float em1 = mrow[jk + 1] ? lrelu(e1r + q1.x) : NEG_INF;
    v2f a;
    a.x = expf(em0 - q0.y) * q0.z;
    a.y = expf(em1 - q1.y) * q1.z;
    v2f b0, b1;
    b0.x = hb[(size_t)jk * F_HID + l];
    b0.y = hb[(size_t)(jk + 1) * F_HID + l];
    b1.x = hb[(size_t)jk * F_HID + 16 + l];
    b1.y = hb[(size_t)(jk + 1) * F_HID + 16 + l];
    c0 = wmma_f32_k4(a, b0, c0);
    c1 = wmma_f32_k4(a, b1, c1);
  }
  float* xr = xcat + (size_t)b * N * HF + (size_t)hh * F_HID;
  for (int r = 0; r < 8; r++) {
    int m = i0 + r + 8 * half;
    xr[(size_t)m * HF + l]      = fmaxf(c0[r], 0.f);
    xr[(size_t)m * HF + 16 + l] = fmaxf(c1[r], 0.f);
  }
}

// -------- h2[b,n,c] = xcat[b,n,:] @ W_out  (one wave = 16 rows x 64 cols) --------
__global__ void __launch_bounds__(256) k_h2(const float* __restrict__ xcat,
                                            const float* __restrict__ Wout,
                                            float* __restrict__ h2) {
  const int lane = threadIdx.x & 31;
  const int wave = blockIdx.x * (blockDim.x >> 5) + (threadIdx.x >> 5);
  const int rt = wave & 31;
  const int b  = wave >> 5;
  const int i0 = rt * 16;
  const int half = lane >> 4;
  const int l = lane & 15;

  const float* xr = xcat + ((size_t)b * N + i0 + l) * HF;
  v8f c0 = {}, c1 = {}, c2 = {}, c3 = {};
  for (int k = 0; k < HF; k += 4) {
    const int kk = k + 2 * half;
    v2f a; a.x = xr[kk]; a.y = xr[kk + 1];
    const float* w0 = Wout + (size_t)kk * C;
    const float* w1 = Wout + (size_t)(kk + 1) * C;
    v2f b0, b1, b2, b3;
    b0.x = w0[l];      b0.y = w1[l];
    b1.x = w0[16 + l]; b1.y = w1[16 + l];
    b2.x = w0[32 + l]; b2.y = w1[32 + l];
    b3.x = w0[48 + l]; b3.y = w1[48 + l];
    c0 = wmma_f32_k4(a, b0, c0);
    c1 = wmma_f32_k4(a, b1, c1);
    c2 = wmma_f32_k4(a, b2, c2);
    c3 = wmma_f32_k4(a, b3, c3);
  }
  for (int r = 0; r < 8; r++) {
    int m = i0 + r + 8 * half;
    float* o = h2 + ((size_t)b * N + m) * C;
    o[l] = c0[r]; o[16 + l] = c1[r]; o[32 + l] = c2[r]; o[48 + l] = c3[r];
  }
}

// -------- e1o/e2o[b,n] = h2[b,n,:] . a_out1/2 --------
__global__ void k_eo(const float* __restrict__ h2, const float* __restrict__ a1,
                     const float* __restrict__ a2, float* __restrict__ e1o,
                     float* __restrict__ e2o) {
  int idx = blockIdx.x * blockDim.x + threadIdx.x;
  if (idx >= B * N) return;
  const float* p = h2 + (size_t)idx * C;
  float s1 = 0.f, s2 = 0.f;
  for (int c = 0; c < C; c++) { float v = p[c]; s1 += v * a1[c]; s2 += v * a2[c]; }
  e1o[idx] = s1; e2o[idx] = s2;
}

// -------- out = log_softmax(elu(P_o @ h2)); P_o built in-flight; LSE via half-wave shuffles ----
__global__ void __launch_bounds__(256) k_final(const unsigned char* __restrict__ mask,
                                               const float* __restrict__ e1o,
                                               const float4* __restrict__ cpko,
                                               const float* __restrict__ h2,
                                               float* __restrict__ out) {
  const int lane = threadIdx.x & 31;
  const int wave = blockIdx.x * (blockDim.x >> 5) + (threadIdx.x >> 5);
  const int rt = wave & 31;
  const int b  = wave >> 5;
  const int i0 = rt * 16;
  const int half = lane >> 4;
  const int l = lane & 15;

  const float e1r = e1o[(size_t)b * N + i0 + l];
  const unsigned char* mrow = mask + (size_t)(i0 + l) * N;
  const float4* cp = cpko + (size_t)b * N;
  const float* hb = h2 + (size_t)b * N * C;

  v8f c0 = {}, c1 = {}, c2 = {}, c3 = {};
  for (int j = 0; j < N; j += 4) {
    const int jk = j + 2 * half;
    float4 q0 = cp[jk];
    float4 q1 = cp[jk + 1];
    float em0 = mrow[jk]     ? lrelu(e1r + q0.x) : NEG_INF;
    float em1 = mrow[jk + 1] ? lrelu(e1r + q1.x) : NEG_INF;
    v2f a;
    a.x = expf(em0 - q0.y) * q0.z;
    a.y = expf(em1 - q1.y) * q1.z;
    const float* r0 = hb + (size_t)jk * C;
    const float* r1 = hb + (size_t)(jk + 1) * C;
    v2f b0, b1, b2, b3;
    b0.x = r0[l];      b0.y = r1[l];
    b1.x = r0[16 + l]; b1.y = r1[16 + l];
    b2.x = r0[32 + l]; b2.y = r1[32 + l];
    b3.x = r0[48 + l]; b3.y = r1[48 + l];
    c0 = wmma_f32_k4(a, b0, c0);
    c1 = wmma_f32_k4(a, b1, c1);
    c2 = wmma_f32_k4(a, b2, c2);
    c3 = wmma_f32_k4(a, b3, c3);
  }
  // Each 16-lane half holds full 64-wide rows: row m = i0 + r + 8*half,
  // cols {l, 16+l, 32+l, 48+l}. elu, then log-softmax over the 64 cols.
  for (int r = 0; r < 8; r++) {
    float v0 = c0[r] > 0.f ? c0[r] : expf(c0[r]) - 1.f;
    float v1 = c1[r] > 0.f ? c1[r] : expf(c1[r]) - 1.f;
    float v2 = c2[r] > 0.f ? c2[r] : expf(c2[r]) - 1.f;
    float v3 = c3[r] > 0.f ? c3[r] : expf(c3[r]) - 1.f;
    float mx = fmaxf(fmaxf(v0, v1), fmaxf(v2, v3));
    for (int s = 1; s < 16; s <<= 1) mx = fmaxf(mx, __shfl_xor(mx, s, 32));
    float sm = expf(v0 - mx) + expf(v1 - mx) + expf(v2 - mx) + expf(v3 - mx);
    for (int s = 1; s < 16; s <<= 1) sm += __shfl_xor(sm, s, 32);
    float lse = mx + logf(sm);
    int m = i0 + r + 8 * half;
    float* o = out + ((size_t)b * N + m) * C;
    o[l] = v0 - lse; o[16 + l] = v1 - lse; o[32 + l] = v2 - lse; o[48 + l] = v3 - lse;
  }
}

extern "C" void kernel_launch(void* const* d_in, const int* in_sizes, int n_in,
                              void* d_out, int out_size, void* d_ws, size_t ws_size,
                              hipStream_t stream) {
  const float* x    = (const float*)d_in[0];
  const float* A    = (const float*)d_in[1];
  const float* W    = (const float*)d_in[2];
  const float* a1   = (const float*)d_in[3];
  const float* a2   = (const float*)d_in[4];
  const float* Wout = (const float*)d_in[5];
  const float* ao1  = (const float*)d_in[6];
  const float* ao2  = (const float*)d_in[7];
  float* out = (float*)d_out;
  (void)in_sizes; (void)n_in; (void)out_size; (void)ws_size;

  char* ws = (char*)d_ws;
  size_t off = 0;
  auto take = [&](size_t bytes) -> char* {
    char* p = ws + off;
    off = (off + bytes + 255) & ~(size_t)255;
    return p;
  };
  unsigned char* mask = (unsigned char*)take((size_t)N * N);           // 256 KB
  float* h     = (float*)take((size_t)B * H * N * F_HID * 4);          // 64 MB
  float* e1    = (float*)take((size_t)B * H * N * 4);                  // 2 MB
  float* e2    = (float*)take((size_t)B * H * N * 4);                  // 2 MB
  float4* cpk  = (float4*)take((size_t)B * H * N * 16);                // 8 MB
  float* xcat  = (float*)take((size_t)B * N * HF * 4);                 // 64 MB
  float* h2    = (float*)take((size_t)B * N * C * 4);                  // 16 MB
  float* e1o   = (float*)take((size_t)B * N * 4);
  float* e2o   = (float*)take((size_t)B * N * 4);
  float4* cpko = (float4*)take((size_t)B * N * 16);

  k_mask<<<(N * N + 255) / 256, 256, 0, stream>>>(A, mask);
  k_h<<<(B * H * (N / 16)) / 8, 256, 0, stream>>>(x, W, h);
  k_e<<<(B * H * N) / 256, 256, 0, stream>>>(h, a1, a2, e1, e2);
  k_colstats<<<B * H, N, 0, stream>>>(mask, e1, e2, cpk);
  k_att<<<(B * H * (N / 16)) / 8, 256, 0, stream>>>(mask, e1, cpk, h, xcat);
  k_h2<<<(B * (N / 16)) / 8, 256, 0, stream>>>(xcat, Wout, h2);
  k_eo<<<(B * N) / 256, 256, 0, stream>>>(h2, ao1, ao2, e1o, e2o);
  k_colstats<<<B, N, 0, stream>>>(mask, e1o, e2o, cpko);
  k_final<<<(B * (N / 16)) / 8, 256, 0, stream>>>(mask, e1o, cpko, h2, out);
}